// GraphAttentionLayer_901943132281
// MI455X (gfx1250) — compile-verified
//
#include <hip/hip_runtime.h>
#include <math.h>

// GAT layer: N=8192, F_in=256, F_out=64, f32 end-to-end (memory-bound problem).
#define N_NODES 8192
#define F_IN    256
#define F_OUT   64
#define LRELU_A 0.2f
#define NEG_BIG (-9.0e15f)

typedef float v2f __attribute__((ext_vector_type(2)));
typedef float v8f __attribute__((ext_vector_type(8)));

__device__ __forceinline__ float wave_reduce_max(float v) {
#pragma unroll
  for (int off = 16; off > 0; off >>= 1)
    v = fmaxf(v, __shfl_xor(v, off, 32));
  return v;
}
__device__ __forceinline__ float wave_reduce_sum(float v) {
#pragma unroll
  for (int off = 16; off > 0; off >>= 1)
    v += __shfl_xor(v, off, 32);
  return v;
}

// ---------------------------------------------------------------------------
// K1: Wh[8192,64] = h[8192,256] @ W[256,64]  via V_WMMA_F32_16X16X4_F32
// Block = 128 threads (4 waves). Each wave owns a 16x16 tile of one 16-row
// stripe; 4 waves cover the 64 output columns. Grid = 8192/16 = 512.
// A-frag layout (ISA 7.12.2, 32-bit A 16x4): lanes 0-15 -> K=0,1 in v0,v1;
// lanes 16-31 -> K=2,3.  B-frag mirrored (row of B striped across lanes).
// ---------------------------------------------------------------------------
__global__ void wh_wmma_kernel(const float* __restrict__ h,
                               const float* __restrict__ W,
                               float* __restrict__ Wh) {
  const int lane = threadIdx.x & 31;
  const int wv   = threadIdx.x >> 5;          // 0..3 -> 16-col tile
  const int m    = lane & 15;
  const int kofs = (lane < 16) ? 0 : 2;
  const int row  = blockIdx.x * 16 + m;
  const int col  = wv * 16 + m;

  v8f acc = {};
  const float* hrow = h + (size_t)row * F_IN;
#pragma unroll 4
  for (int k = 0; k < F_IN; k += 4) {
    v2f af, bf;
    af.x = hrow[k + kofs];
    af.y = hrow[k + kofs + 1];
    bf.x = W[(size_t)(k + kofs)     * F_OUT + col];
    bf.y = W[(size_t)(k + kofs + 1) * F_OUT + col];
    acc = __builtin_amdgcn_wmma_f32_16x16x4_f32(false, af, false, bf,
                                                (short)0, acc, false, false);
  }
  // D layout: VGPR r -> M=r (lanes 0-15) / M=r+8 (lanes 16-31), N = lane&15
  const int rbase = blockIdx.x * 16 + ((lane >= 16) ? 8 : 0);
#pragma unroll
  for (int r = 0; r < 8; ++r)
    Wh[(size_t)(rbase + r) * F_OUT + col] = acc[r];
}

// ---------------------------------------------------------------------------
// K2: s1[i] = Wh[i,:] . a[0:64],  s2[i] = Wh[i,:] . a[64:128]
// One wave32 per row; shuffle reduction. Grid = 8192/8 blocks of 256.
// ---------------------------------------------------------------------------
__global__ void scores_kernel(const float* __restrict__ Wh,
                              const float* __restrict__ a,
                              float* __restrict__ s1,
                              float* __restrict__ s2) {
  const int lane = threadIdx.x & 31;
  const int wv   = threadIdx.x >> 5;
  const int row  = blockIdx.x * 8 + wv;
  const float w0 = Wh[(size_t)row * F_OUT + lane];
  const float w1 = Wh[(size_t)row * F_OUT + lane + 32];
  float d1 = w0 * a[lane]         + w1 * a[lane + 32];
  float d2 = w0 * a[F_OUT + lane] + w1 * a[F_OUT + lane + 32];
  d1 = wave_reduce_sum(d1);
  d2 = wave_reduce_sum(d2);
  if (lane == 0) { s1[row] = d1; s2[row] = d2; }
}

// ---------------------------------------------------------------------------
// K3: row-wise masked softmax. One 256-thread block per row.
// e_ij = leakyrelu(s1[i]+s2[j]); masked by adj; two-pass softmax with the
// full row (32 KB) staged in LDS. Writes attention row (coalesced).
// ---------------------------------------------------------------------------
__global__ void softmax_row_kernel(const int*   __restrict__ adj,
                                   const float* __restrict__ s1,
                                   const float* __restrict__ s2,
                                   float* __restrict__ att) {
  __shared__ float e[N_NODES];   // 32 KB of the 320 KB/WGP LDS
  __shared__ float redmax[8];
  __shared__ float redsum[8];
  const int tid  = threadIdx.x;
  const int lane = tid & 31;
  const int wv   = tid >> 5;
  const int row  = blockIdx.x;
  const float s1i = s1[row];
  const int* arow = adj + (size_t)row * N_NODES;

  // pass 1: masked leaky-relu scores + row max
  float m = -INFINITY;
  for (int j = tid; j < N_NODES; j += 256) {
    float v = s1i + s2[j];
    v = (v >= 0.0f) ? v : LRELU_A * v;
    v = (arow[j] > 0) ? v : NEG_BIG;
    e[j] = v;
    m = fmaxf(m, v);
  }
  m = wave_reduce_max(m);
  if (lane == 0) redmax[wv] = m;
  __syncthreads();
  float M = redmax[0];
#pragma unroll
  for (int i = 1; i < 8; ++i) M = fmaxf(M, redmax[i]);

  // pass 2: exp + row sum (each thread touches only its own e[j] slots)
  float s = 0.0f;
  for (int j = tid; j < N_NODES; j += 256) {
    const float p = __expf(e[j] - M);
    e[j] = p;
    s += p;
  }
  s = wave_reduce_sum(s);
  if (lane == 0) redsum[wv] = s;
  __syncthreads();
  float S = 0.0f;
#pragma unroll
  for (int i = 0; i < 8; ++i) S += redsum[i];
  const float inv = 1.0f / S;

  // pass 3: normalized write-out
  float* orow = att + (size_t)row * N_NODES;
  for (int j = tid; j < N_NODES; j += 256)
    orow[j] = e[j] * inv;
}

// ---------------------------------------------------------------------------
// K4: out = ELU( attention[8192,8192] @ Wh[8192,64] ) via WMMA f32 16x16x4.
// Block = 256 threads (8 waves) -> 32 rows x 64 cols; wave (rw,cw) owns a
// 16x16 tile. K-chunks of 64 staged through LDS (As padded: stride-64 f32
// would land all 16 lanes on one of the 64 LDS banks).
// Wh (2 MB) stays L2-resident across the 256 blocks.
// ---------------------------------------------------------------------------
#define TK 64
__global__ void av_gemm_kernel(const float* __restrict__ att,
                               const float* __restrict__ Wh,
                               float* __restrict__ out) {
  __shared__ float As[32][TK + 4];   // +4 pad kills bank conflicts on A reads
  __shared__ float Bs[TK][F_OUT];
  const int tid  = threadIdx.x;
  const int lane = tid & 31;
  const int wv   = tid >> 5;           // 0..7
  const int rw   = wv >> 2;            // 0..1  (16-row sub-tile)
  const int cw   = wv & 3;             // 0..3  (16-col tile)
  const int m    = lane & 15;
  const int kofs = (lane < 16) ? 0 : 2;
  const int rowbase = blockIdx.x * 32;

  v8f acc = {};
  for (int kb = 0; kb < N_NODES; kb += TK) {
    // stage A tile: 32 x 64 f32, float4 per thread x2, coalesced along k
#pragma unroll
    for (int t = 0; t < 2; ++t) {
      const int lin = (tid + t * 256) * 4;     // 0..2047
      const int r  = lin >> 6;
      const int kk = lin & 63;
      const float4 v =
          *(const float4*)(att + (size_t)(rowbase + r) * N_NODES + kb + kk);
      As[r][kk] = v.x; As[r][kk + 1] = v.y;
      As[r][kk + 2] = v.z; As[r][kk + 3] = v.w;
    }
    // stage B tile: 64 x 64 f32
#pragma unroll
    for (int t = 0; t < 4; ++t) {
      const int lin = (tid + t * 256) * 4;     // 0..4095
      const int r = lin >> 6;
      const int c = lin & 63;
      *(float4*)(&Bs[r][c]) =
          *(const float4*)(Wh + (size_t)(kb + r) * F_OUT + c);
    }
    __syncthreads();
#pragma unroll
    for (int k2 = 0; k2 < TK; k2 += 4) {
      v2f af, bf;
      af.x = As[rw * 16 + m][k2 + kofs];
      af.y = As[rw * 16 + m][k2 + kofs + 1];
      bf.x = Bs[k2 + kofs][cw * 16 + m];
      bf.y = Bs[k2 + kofs + 1][cw * 16 + m];
      acc = __builtin_amdgcn_wmma_f32_16x16x4_f32(false, af, false, bf,
                                                  (short)0, acc, false, false);
    }
    __syncthreads();
  }
  const int row0 = rowbase + rw * 16 + ((lane >= 16) ? 8 : 0);
  const int col  = cw * 16 + m;
#pragma unroll
  for (int r = 0; r < 8; ++r) {
    float x = acc[r];
    x = (x > 0.0f) ? x : expm1f(x);   // ELU, alpha=1
    out[(size_t)(row0 + r) * F_OUT + col] = x;
  }
}

// ---------------------------------------------------------------------------
extern "C" void kernel_launch(void* const* d_in, const int* in_sizes, int n_in,
                              void* d_out, int out_size, void* d_ws, size_t ws_size,
                              hipStream_t stream) {
  const float* h   = (const float*)d_in[0];   // [8192, 256]
  const int*   adj = (const int*)  d_in[1];   // [8192, 8192]
  const float* W   = (const float*)d_in[2];   // [256, 64]
  const float* a   = (const float*)d_in[3];   // [128, 1]

  float* out = (float*)d_out;                          // [8192, 64]
  float* att = out + (size_t)N_NODES * F_OUT;          // [8192, 8192]

  float* Wh = (float*)d_ws;                            // 2 MB
  float* s1 = Wh + (size_t)N_NODES * F_OUT;            // 32 KB
  float* s2 = s1 + N_NODES;                            // 32 KB

  wh_wmma_kernel     <<<N_NODES / 16, 128, 0, stream>>>(h, W, Wh);
  scores_kernel      <<<N_NODES / 8,  256, 0, stream>>>(Wh, a, s1, s2);
  softmax_row_kernel <<<N_NODES,      256, 0, stream>>>(adj, s1, s2, att);
  av_gemm_kernel     <<<N_NODES / 32, 256, 0, stream>>>(att, Wh, out);
}